// ScoreModel_12652973654329
// MI455X (gfx1250) — compile-verified
//
#include <hip/hip_runtime.h>
#include <math.h>

typedef float v2f __attribute__((ext_vector_type(2)));
typedef float v8f __attribute__((ext_vector_type(8)));

#define N_T    64
#define N_Q    1024
#define N_K    1024
#define D_FEAT 160
#define MUL1   32
#define KP_COLS 128   // compact key_proj row: [0:32]=scalars 0..31, [32:128]=32 vec3 (cols 64..159)

// ---- CDNA5 async global->LDS DMA (ASYNCcnt-tracked), portable inline asm ----
__device__ __forceinline__ void async_copy_b128(unsigned lds_byte_off, const void* gaddr) {
    asm volatile("global_load_async_to_lds_b128 %0, %1, off"
                 :: "v"(lds_byte_off), "v"(gaddr) : "memory");
}
__device__ __forceinline__ void wait_asynccnt0() {
    asm volatile("s_wait_asynccnt 0x0" ::: "memory");
}

// ---------------- kernel 0: zero the output accumulator ----------------
__global__ void zero_out_kernel(float* __restrict__ out, int n) {
    int i = blockIdx.x * blockDim.x + threadIdx.x;
    if (i < n) out[i] = 0.0f;
}

// ---------------- kernel 1: wbar = row-mean of Wv (96x32 -> 96) --------
__global__ void wbar_kernel(const float* __restrict__ wv_lin,
                            const float* __restrict__ wv_ang,
                            float* __restrict__ wbar) {
    int i = threadIdx.x;
    if (i >= 192) return;
    const float* src = (i < 96) ? (wv_lin + i * 32) : (wv_ang + (i - 96) * 32);
    float s = 0.0f;
#pragma unroll
    for (int o = 0; o < 32; ++o) s += src[o];
    wbar[i] = s * (1.0f / 32.0f);
}

// ---------------- kernel 2: key_proj = key_feat @ W_key (WMMA f32) -----
// A: 1024x160 row-major, B: 160x160 row-major.
// Output compact: 1024 x 128 (cols {0..31} U {64..159}).
// One wave per 16x16 C tile; 64 M-tiles x 8 N-tiles = 512 waves.
__global__ __launch_bounds__(128)
void kproj_gemm_kernel(const float* __restrict__ A,
                       const float* __restrict__ B,
                       float* __restrict__ C) {
    const int lane = threadIdx.x & 31;
    const int wave = threadIdx.x >> 5;
    const int gw   = blockIdx.x * 4 + wave;   // 0..511
    const int tm   = gw >> 3;                 // 0..63
    const int tn   = gw & 7;                  // 0..7
    const int colSel = (tn < 2) ? tn * 16 : 64 + (tn - 2) * 16;

    const int row   = tm * 16 + (lane & 15);        // A row for this lane
    const int colB  = colSel + (lane & 15);         // B col for this lane
    const int khalf = (lane >> 4) << 1;             // lanes 16..31 own K+2,K+3

    v8f acc = {};
#pragma unroll 4
    for (int k0 = 0; k0 < D_FEAT; k0 += 4) {
        const int ka = k0 + khalf;
        v2f a, b;
        // A frag (16x4): lane<16 -> (K=k0,k0+1), lane>=16 -> (K=k0+2,k0+3)
        a.x = A[row * D_FEAT + ka];
        a.y = A[row * D_FEAT + ka + 1];
        // B frag (4x16): same K split, N striped across lanes 0..15
        b.x = B[ka * D_FEAT + colB];
        b.y = B[(ka + 1) * D_FEAT + colB];
        acc = __builtin_amdgcn_wmma_f32_16x16x4_f32(false, a, false, b,
                                                    (short)0, acc, false, false);
    }
    // C layout: VGPR r -> rows (r | r+8), lanes 0..15 / 16..31; cols = lane&15
    const int cc    = tn * 16 + (lane & 15);
    const int rbase = tm * 16 + ((lane >> 4) ? 8 : 0);
#pragma unroll
    for (int r = 0; r < 8; ++r)
        C[(rbase + r) * KP_COLS + cc] = acc[r];
}

// ---------------- kernel 3: fused main kernel --------------------------
// One thread per (t,q). Per block: t fixed, 256 q's.
__global__ __launch_bounds__(256)
void score_main_kernel(const float* __restrict__ T,
                       const float* __restrict__ qwgt,
                       const float* __restrict__ qfeat,
                       const float* __restrict__ qcoord,
                       const float* __restrict__ kcoord,
                       const float* __restrict__ kproj,
                       const float* __restrict__ wbar,
                       float* __restrict__ out) {
    __shared__ __align__(16) float sKC3[N_K * 3];  // 12 KB packed xyz (raw async copy)
    __shared__ __align__(16) float sW[192];        // wbar_lin(96) + wbar_ang(96)
    __shared__ float sRed[8][6];                   // per-wave partials

    const int tid = threadIdx.x;

    // ---- async DMA staging: global -> LDS, tracked by ASYNCcnt ----
    {
        const unsigned ldsKC = (unsigned)(size_t)(&sKC3[0]);
        const char* gk = (const char*)kcoord;          // 12288 bytes = 768 x b128
        for (int i = tid; i < 768; i += 256)
            async_copy_b128(ldsKC + (unsigned)i * 16u, gk + (size_t)i * 16u);
        const unsigned ldsW = (unsigned)(size_t)(&sW[0]);
        if (tid < 48)                                   // 768 bytes = 48 x b128
            async_copy_b128(ldsW + (unsigned)tid * 16u, (const char*)wbar + (size_t)tid * 16u);
        wait_asynccnt0();
    }
    __syncthreads();

    const int t = blockIdx.y;
    const int q = blockIdx.x * 256 + tid;

    // ---- normalized quaternion -> rotation matrix R, translation X ----
    float qw0 = T[t*7+0], qx = T[t*7+1], qy = T[t*7+2], qz = T[t*7+3];
    float inv = rsqrtf(qw0*qw0 + qx*qx + qy*qy + qz*qz);
    qw0 *= inv; qx *= inv; qy *= inv; qz *= inv;
    const float Xx = T[t*7+4], Xy = T[t*7+5], Xz = T[t*7+6];
    const float r00 = 1.0f - 2.0f*(qy*qy + qz*qz);
    const float r01 = 2.0f*(qx*qy - qw0*qz);
    const float r02 = 2.0f*(qx*qz + qw0*qy);
    const float r10 = 2.0f*(qx*qy + qw0*qz);
    const float r11 = 1.0f - 2.0f*(qx*qx + qz*qz);
    const float r12 = 2.0f*(qy*qz - qw0*qx);
    const float r20 = 2.0f*(qx*qz - qw0*qy);
    const float r21 = 2.0f*(qy*qz + qw0*qx);
    const float r22 = 1.0f - 2.0f*(qx*qx + qy*qy);

    // ---- transformed query point p = R*c + X ----
    const float cx = qcoord[3*q], cy = qcoord[3*q+1], cz = qcoord[3*q+2];
    const float px = r00*cx + r01*cy + r02*cz + Xx;
    const float py = r10*cx + r11*cy + r12*cz + Xy;
    const float pz = r20*cx + r21*cy + r22*cz + Xz;

    // ---- nearest-neighbor over 1024 keys in LDS (first-min semantics) ----
    float best = 3.402823466e38f;
    int   bidx = 0;
#pragma unroll 4
    for (int k = 0; k < N_K; ++k) {
        const float dx = px - sKC3[3*k];
        const float dy = py - sKC3[3*k+1];
        const float dz = pz - sKC3[3*k+2];
        const float d2 = fmaf(dx, dx, fmaf(dy, dy, dz*dz));
        if (d2 < best) { best = d2; bidx = k; }
    }
    const float env = expf(-sqrtf(best));

    const float* kp = kproj + bidx * KP_COLS;
    const float* qf = qfeat + q * D_FEAT;
    __builtin_prefetch(kp, 0, 0);           // global_prefetch_b8: pull gathered row
    __builtin_prefetch(kp + 64, 0, 0);      // second cacheline of the 512B row

    // ---- 32-channel irrep contraction (in query frame, env folded in) ----
    float lx = 0.f, ly = 0.f, lz = 0.f;
    float ax = 0.f, ay = 0.f, az = 0.f;
#pragma unroll 4
    for (int c = 0; c < MUL1; ++c) {
        const float sbv = env * kp[c];
        const float bx0 = kp[32 + 3*c], by0 = kp[32 + 3*c + 1], bz0 = kp[32 + 3*c + 2];
        // vb' = env * R^T * vb   (back-rotate key vectors once)
        const float vbx = env * (r00*bx0 + r10*by0 + r20*bz0);
        const float vby = env * (r01*bx0 + r11*by0 + r21*bz0);
        const float vbz = env * (r02*bx0 + r12*by0 + r22*bz0);
        const float sav = qf[c];
        const float vax = qf[64 + 3*c], vay = qf[64 + 3*c + 1], vaz = qf[64 + 3*c + 2];
        const float crx = vay*vbz - vaz*vby;
        const float cry = vaz*vbx - vax*vbz;
        const float crz = vax*vby - vay*vbx;
        const float a1l = sW[c]       * sav;
        const float a2l = sW[32 + c]  * sbv;
        const float a3l = sW[64 + c];
        const float a1a = sW[96 + c]  * sav;
        const float a2a = sW[128 + c] * sbv;
        const float a3a = sW[160 + c];
        lx = fmaf(a1l, vbx, fmaf(a2l, vax, fmaf(a3l, crx, lx)));
        ly = fmaf(a1l, vby, fmaf(a2l, vay, fmaf(a3l, cry, ly)));
        lz = fmaf(a1l, vbz, fmaf(a2l, vaz, fmaf(a3l, crz, lz)));
        ax = fmaf(a1a, vbx, fmaf(a2a, vax, fmaf(a3a, crx, ax)));
        ay = fmaf(a1a, vby, fmaf(a2a, vay, fmaf(a3a, cry, ay)));
        az = fmaf(a1a, vbz, fmaf(a2a, vaz, fmaf(a3a, crz, az)));
    }

    // ---- ang_orbital = cross(query_coord, lin); weight by query_weight ----
    const float obx = cy*lz - cz*ly;
    const float oby = cz*lx - cx*lz;
    const float obz = cx*ly - cy*lx;
    const float wq = qwgt[q];
    float vals[6] = { wq*lx, wq*ly, wq*lz,
                      wq*(ax + obx), wq*(ay + oby), wq*(az + obz) };

    // ---- wave32 shuffle reduce, then cross-wave via LDS, then atomics ----
#pragma unroll
    for (int j = 0; j < 6; ++j) {
        float v = vals[j];
        for (int off = 16; off > 0; off >>= 1) v += __shfl_down(v, off, 32);
        vals[j] = v;
    }
    const int wv = tid >> 5, ln = tid & 31;
    if (ln == 0) {
#pragma unroll
        for (int j = 0; j < 6; ++j) sRed[wv][j] = vals[j];
    }
    __syncthreads();
    if (tid < 6) {
        float s = 0.0f;
#pragma unroll
        for (int w = 0; w < 8; ++w) s += sRed[w][tid];
        const int comp = (tid < 3) ? tid : tid - 3;
        const int base = (tid < 3) ? 0 : (N_T * 3);     // lin_vel then ang_vel
        atomicAdd(&out[base + t*3 + comp], s);
    }
}

// ---------------- launcher --------------------------------------------
extern "C" void kernel_launch(void* const* d_in, const int* in_sizes, int n_in,
                              void* d_out, int out_size, void* d_ws, size_t ws_size,
                              hipStream_t stream) {
    const float* T      = (const float*)d_in[0];   // 64x7
    const float* qwgt   = (const float*)d_in[1];   // 1024
    const float* qfeat  = (const float*)d_in[2];   // 1024x160
    const float* qcoord = (const float*)d_in[3];   // 1024x3
    const float* kcoord = (const float*)d_in[4];   // 1024x3
    const float* kfeat  = (const float*)d_in[5];   // 1024x160
    const float* Wkey   = (const float*)d_in[6];   // 160x160
    // d_in[7] = Ws_lin (unused: scalar dtp output is discarded)
    const float* Wv_lin = (const float*)d_in[8];   // 96x32
    // d_in[9] = Ws_ang (unused)
    const float* Wv_ang = (const float*)d_in[10];  // 96x32

    float* out = (float*)d_out;                    // 384 = lin_vel(64x3)+ang_vel(64x3)

    // workspace: key_proj compact (1024*128 f32) then wbar (192 f32)
    float* kp   = (float*)d_ws;
    float* wbar = kp + (size_t)N_K * KP_COLS;

    zero_out_kernel<<<(out_size + 255) / 256, 256, 0, stream>>>(out, out_size);
    wbar_kernel<<<1, 192, 0, stream>>>(Wv_lin, Wv_ang, wbar);
    kproj_gemm_kernel<<<128, 128, 0, stream>>>(kfeat, Wkey, kp);
    score_main_kernel<<<dim3(N_Q / 256, N_T), 256, 0, stream>>>(
        T, qwgt, qfeat, qcoord, kcoord, kp, wbar, out);
}